// RNN_74620761800895
// MI455X (gfx1250) — compile-verified
//
#include <hip/hip_runtime.h>

typedef float v2f __attribute__((ext_vector_type(2)));
typedef float v8f __attribute__((ext_vector_type(8)));
typedef unsigned int u32x4 __attribute__((ext_vector_type(4)));
typedef int          i32x4 __attribute__((ext_vector_type(4)));
typedef int          i32x8 __attribute__((ext_vector_type(8)));

constexpr int B_   = 1024;
constexpr int T_   = 256;
constexpr int IN_  = 256;
constexpr int H_   = 128;
constexpr int OUT_ = 64;
constexpr int C_   = IN_ + H_;   // 384 concat width
constexpr int LW   = C_ + 4;     // 388: weight row stride (floats), conflict-free
constexpr int LX   = IN_ + 4;    // 260: x-tile row stride
constexpr int LH   = H_ + 4;     // 132: hidden row stride

// Manual LDS layout (floats): x buffers at 0 so DS immediate offsets apply.
constexpr int XOFF = 0;                 // ldsX[2][16*LX]  (2 x 16640 B)
constexpr int HOFF = 2 * 16 * LX;       // ldsH[16*LH]     (base 33280 B)
constexpr int WOFF = HOFF + 16 * LH;    // ldsW[H_*LW]     (base 41728 B)
constexpr int LDS_TOTAL = WOFF + H_ * LW;   // 60096 floats = 240384 B < 320 KB

// Generic shared pointer -> 32-bit LDS byte offset (flat LDS addr = low 32 bits).
__device__ __forceinline__ unsigned lds_off(const void* p) {
  return (unsigned)(unsigned long long)(uintptr_t)p;
}

// Issue one TDM descriptor: copy a 16-row x 256-float tile (row stride T*IN
// floats in memory) into LDS at lds_byte_off, inserting 4 dwords of padding
// after every 256 dwords stored (-> LDS row stride 260 floats).
__device__ __forceinline__ void tdm_load_x_tile(unsigned lds_byte_off,
                                                unsigned long long gaddr) {
  u32x4 g0 = { 1u,                                   // count=1 valid descriptor
               lds_byte_off,                         // lds_addr
               (unsigned)gaddr,                      // global_addr[31:0]
               0x80000000u | (unsigned)(gaddr >> 32) // global_addr[56:32] | type=2
             };
  i32x8 g1 = { (int)((2u << 16)    // data_size = 4 bytes
                   | (1u << 20)    // pad_enable
                   | (7u << 22)    // pad_interval = 256 dwords (one row)
                   | (3u << 25)),  // pad_amount = 4 dwords
               (int)(IN_ << 16),   // tensor_dim0[15:0] = 256    (bits 79:48)
               (int)(16  << 16),   // tensor_dim1[15:0] = 16     (bits 111:80)
               (int)(IN_ << 16),   // tile_dim0 = 256            (bits 127:112)
               16,                 // tile_dim1 = 16 rows
               (int)(T_ * IN_),    // tensor_dim0_stride = 65536 elements
               0, 0 };
  i32x4 z4 = { 0, 0, 0, 0 };               // groups 2/3 unused (2-D tensor)
  i32x8 z8 = { 0, 0, 0, 0, 0, 0, 0, 0 };
  __builtin_amdgcn_tensor_load_to_lds(g0, g1, z4, z4, z8, 0);
}

// One block = 16 batch rows; full 255-step recurrence runs locally (batch rows
// are independent). 8 waves x 16 output columns, fp32 WMMA 16x16x4. x tiles are
// streamed by the Tensor Data Mover, double-buffered against the WMMA chain.
// The per-wave Wh slice (hidden-part weights) is register-resident.
__global__ __launch_bounds__(256) void rnn_fused_kernel(
    const float* __restrict__ x,    // [B, T, IN]
    const float* __restrict__ Wih,  // [H, C]
    const float* __restrict__ bih,  // [H]
    const float* __restrict__ Wio,  // [OUT, C]
    const float* __restrict__ bio,  // [OUT]
    float* __restrict__ out)        // [B, OUT]
{
  __shared__ __align__(16) float lds[LDS_TOTAL];
  float* ldsX = lds + XOFF;   // [2][16 * LX]
  float* ldsH = lds + HOFF;   // [16 * LH]
  float* ldsW = lds + WOFF;   // [H_ * LW]

  const int tid  = threadIdx.x;
  const int wave = tid >> 5;         // 0..7 (wave32)
  const int lane = tid & 31;
  const int mla  = lane & 15;        // M (or N) index within 16
  const int hi   = lane >> 4;        // lane half: K-pair select / M+8 rows
  const int koff = hi << 1;          // lanes 16..31 hold K+2,K+3
  const int b0   = blockIdx.x << 4;  // first batch row of this block

  const unsigned long long xbase =
      (unsigned long long)(uintptr_t)x +
      (unsigned long long)b0 * T_ * IN_ * 4ull;

  // Kick off the TDM for t=0 immediately (overlaps the W staging below).
  if (wave == 0)
    tdm_load_x_tile(lds_off(ldsX), xbase);

  // ---- Stage W_i2h [128 x 384] into LDS once ----
  for (int idx = tid; idx < H_ * (C_ / 4); idx += 256) {
    int row = idx / (C_ / 4);
    int c4  = idx - row * (C_ / 4);
    float4 v = *(const float4*)(Wih + (size_t)row * C_ + c4 * 4);
    *(float4*)(&ldsW[row * LW + c4 * 4]) = v;
  }
  // ---- hidden := 0 ----
  for (int idx = tid; idx < 16 * H_; idx += 256) {
    int row = idx >> 7;
    int cc  = idx & (H_ - 1);
    ldsH[row * LH + cc] = 0.0f;
  }
  __syncthreads();  // W staged before register preload below

  const int   n0   = wave << 4;           // this wave's output-column tile
  const float biah = bih[n0 + mla];

  // B operand: W[N=n0+mla][K] == (W^T)[K][N]; A operands follow ISA 16x4 f32 layout.
  const float* Bb  = &ldsW[(n0 + mla) * LW + koff];
  const float* AbH = &ldsH[mla * LH + koff];

  // Hoist the loop-invariant hidden-segment weights (K=256..383) into VGPRs:
  // 32 x v2f = 64 VGPRs, reused by all 255 timesteps.
  v2f wH[32];
  #pragma unroll
  for (int i = 0; i < 16; ++i) {
    wH[2 * i]     = *(const v2f*)(Bb + IN_ + i * 8);
    wH[2 * i + 1] = *(const v2f*)(Bb + IN_ + i * 8 + 4);
  }

  for (int t = 0; t < T_ - 1; ++t) {
    if (wave == 0) {
      // Start DMA of the NEXT timestep's tile into the other buffer, then
      // wait until only that one is outstanding => tile t has landed in LDS.
      tdm_load_x_tile(lds_off(ldsX + ((t + 1) & 1) * (16 * LX)),
                      xbase + (unsigned long long)(t + 1) * IN_ * 4ull);
      __builtin_amdgcn_s_wait_tensorcnt(1);
    }
    __syncthreads();  // x[t] tile + previous hidden visible to all waves

    const float* AbX = ldsX + (t & 1) * (16 * LX) + mla * LX + koff;
    v8f c0 = {}, c1 = {};
    #pragma unroll 4
    for (int k = 0; k < IN_; k += 8) {        // x segment: K = 0..255
      v2f a0 = *(const v2f*)(AbX + k);
      v2f w0 = *(const v2f*)(Bb + k);
      c0 = __builtin_amdgcn_wmma_f32_16x16x4_f32(false, a0, false, w0,
                                                 (short)0, c0, false, false);
      v2f a1 = *(const v2f*)(AbX + k + 4);
      v2f w1 = *(const v2f*)(Bb + k + 4);
      c1 = __builtin_amdgcn_wmma_f32_16x16x4_f32(false, a1, false, w1,
                                                 (short)0, c1, false, false);
    }
    #pragma unroll
    for (int i = 0; i < 16; ++i) {            // hidden segment: K = 256..383
      v2f a0 = *(const v2f*)(AbH + i * 8);
      c0 = __builtin_amdgcn_wmma_f32_16x16x4_f32(false, a0, false, wH[2 * i],
                                                 (short)0, c0, false, false);
      v2f a1 = *(const v2f*)(AbH + i * 8 + 4);
      c1 = __builtin_amdgcn_wmma_f32_16x16x4_f32(false, a1, false, wH[2 * i + 1],
                                                 (short)0, c1, false, false);
    }
    __syncthreads();  // everyone finished reading ldsH before overwrite

    v8f c = c0 + c1;
    #pragma unroll
    for (int r = 0; r < 8; ++r)               // C/D layout: M = r + 8*hi, N = mla
      ldsH[(r + (hi << 3)) * LH + n0 + mla] = c[r] + biah;
  }

  // ---- Final projection: out = [x_{T-1} | h] @ W_i2o^T + b_i2o ----
  for (int idx = tid; idx < OUT_ * (C_ / 4); idx += 256) {
    int row = idx / (C_ / 4);
    int c4  = idx - row * (C_ / 4);
    float4 v = *(const float4*)(Wio + (size_t)row * C_ + c4 * 4);
    *(float4*)(&ldsW[row * LW + c4 * 4]) = v;
  }
  if (wave == 0)
    __builtin_amdgcn_s_wait_tensorcnt(0);     // x[T-1] tile (issued at t=254)
  __syncthreads();

  if (wave < 4) {                             // OUT = 64 -> 4 column tiles
    const float bo = bio[n0 + mla];
    const float* AbX = ldsX + ((T_ - 1) & 1) * (16 * LX) + mla * LX + koff;
    v8f c0 = {}, c1 = {};
    #pragma unroll 4
    for (int k = 0; k < IN_; k += 8) {
      v2f a0 = *(const v2f*)(AbX + k);
      v2f w0 = *(const v2f*)(Bb + k);
      c0 = __builtin_amdgcn_wmma_f32_16x16x4_f32(false, a0, false, w0,
                                                 (short)0, c0, false, false);
      v2f a1 = *(const v2f*)(AbX + k + 4);
      v2f w1 = *(const v2f*)(Bb + k + 4);
      c1 = __builtin_amdgcn_wmma_f32_16x16x4_f32(false, a1, false, w1,
                                                 (short)0, c1, false, false);
    }
    #pragma unroll 4
    for (int k = 0; k < H_; k += 8) {         // W_i2o hidden segment from LDS
      v2f a0 = *(const v2f*)(AbH + k);
      v2f w0 = *(const v2f*)(Bb + IN_ + k);
      c0 = __builtin_amdgcn_wmma_f32_16x16x4_f32(false, a0, false, w0,
                                                 (short)0, c0, false, false);
      v2f a1 = *(const v2f*)(AbH + k + 4);
      v2f w1 = *(const v2f*)(Bb + IN_ + k + 4);
      c1 = __builtin_amdgcn_wmma_f32_16x16x4_f32(false, a1, false, w1,
                                                 (short)0, c1, false, false);
    }
    v8f c = c0 + c1;
    #pragma unroll
    for (int r = 0; r < 8; ++r)
      out[(size_t)(b0 + r + (hi << 3)) * OUT_ + n0 + mla] = c[r] + bo;
  }
}

extern "C" void kernel_launch(void* const* d_in, const int* in_sizes, int n_in,
                              void* d_out, int out_size, void* d_ws, size_t ws_size,
                              hipStream_t stream) {
  (void)in_sizes; (void)n_in; (void)out_size; (void)d_ws; (void)ws_size;
  const float* x   = (const float*)d_in[0];
  const float* Wih = (const float*)d_in[1];
  const float* bih = (const float*)d_in[2];
  const float* Wio = (const float*)d_in[3];
  const float* bio = (const float*)d_in[4];
  float* out = (float*)d_out;
  rnn_fused_kernel<<<dim3(B_ / 16), dim3(256), 0, stream>>>(x, Wih, bih, Wio, bio, out);
}